// Adaptive_Margin_Loss_6786048328134
// MI455X (gfx1250) — compile-verified
//
#include <hip/hip_runtime.h>

typedef __attribute__((ext_vector_type(2))) float v2f;
typedef __attribute__((ext_vector_type(8))) float v8f;

#define B_DIM 256
#define C_DIM 4096
#define D_DIM 100
#define NBLK  8192   // 65536 wave-tiles / 8 waves per block

__device__ __forceinline__ float block_reduce_sum(float v, float* sm) {
    const int t = threadIdx.x;
    sm[t] = v;
    __syncthreads();
    #pragma unroll
    for (int s = 128; s > 0; s >>= 1) {
        if (t < s) sm[t] += sm[t + s];
        __syncthreads();
    }
    return sm[0];
}

// loss1 = sum over B*D of (pos - lan)^2  -> ws[0]
__global__ void __launch_bounds__(256) loss1_kernel(const float* __restrict__ pos,
                                                    const float* __restrict__ lan,
                                                    float* __restrict__ ws) {
    __shared__ float sm[256];
    float s = 0.f;
    for (int i = threadIdx.x; i < B_DIM * D_DIM; i += 256) {
        float d = pos[i] - lan[i];
        s += d * d;
    }
    float tot = block_reduce_sum(s, sm);
    if (threadIdx.x == 0) ws[0] = tot;
}

// Each wave: one (b, c0..c0+15) tile. diff = neg_tile - pos (16x4 per step).
// acc = diff * diff^T accumulated over 25 K=4 chunks (D=100).
// diag(acc)[m] = sum_d diff[m,d]^2. relu(margin - diag/100), block-reduce.
__global__ void __launch_bounds__(256) loss2_kernel(const float* __restrict__ pos,
                                                    const float* __restrict__ neg,
                                                    float* __restrict__ ws) {
    __shared__ float sm[256];
    const int lane = threadIdx.x & 31;
    const int wave = threadIdx.x >> 5;
    const int tile = blockIdx.x * 8 + wave;       // 0 .. 65535
    const int b    = tile >> 8;                   // tile / (C/16)
    const int c0   = (tile & 255) << 4;           // 16-wide c tile
    const int row  = lane & 15;                   // A-matrix row (c index)
    const int koff = (lane >> 4) << 1;            // lanes 0-15: K=0,1 ; 16-31: K=2,3

    const float* __restrict__ np =
        neg + ((size_t)b * C_DIM + (size_t)(c0 + row)) * D_DIM + koff;
    const float* __restrict__ pp = pos + (size_t)b * D_DIM + koff;

    v8f acc = {};
    #pragma unroll
    for (int j = 0; j < 25; ++j) {
        const int d0 = j * 4;
        float2 nv = *(const float2*)(np + d0);    // 8B aligned (all offsets even)
        float2 pv = *(const float2*)(pp + d0);
        v2f a;
        a.x = nv.x - pv.x;
        a.y = nv.y - pv.y;
        // B register layout for 4x16 f32 mirrors A's 16x4 layout, so passing
        // the same registers computes D = A * A^T (Gram matrix of diff tile).
        acc = __builtin_amdgcn_wmma_f32_16x16x4_f32(
            /*neg_a=*/false, a, /*neg_b=*/false, a,
            /*c_mod=*/(short)0, acc, /*reuse_a=*/false, /*reuse_b=*/false);
    }

    // Diagonal of C/D 16x16 f32 layout: VGPR r holds M=r (lanes 0-15, N=lane)
    // and M=r+8 (lanes 16-31, N=lane-16). M==N lives at lane r (VGPR r) and
    // lane r+24 (VGPR r).
    const int idx = (lane < 8) ? lane : ((lane >= 24) ? (lane - 24) : -1);
    float dsum = 0.f;
    #pragma unroll
    for (int r = 0; r < 8; ++r) dsum = (idx == r) ? acc[r] : dsum;

    float val = 0.f;
    if (idx >= 0) {
        float d2 = dsum * (1.0f / (float)D_DIM);
        float h  = 0.1f - d2;                     // MARGIN = 0.1
        val = h > 0.f ? h : 0.f;
    }

    float bs = block_reduce_sum(val, sm);
    if (threadIdx.x == 0) ws[1 + blockIdx.x] = bs;
}

__global__ void __launch_bounds__(256) finalize_kernel(const float* __restrict__ ws,
                                                       float* __restrict__ out) {
    __shared__ float sm[256];
    float s = 0.f;
    for (int i = threadIdx.x; i < NBLK; i += 256) s += ws[1 + i];
    float tot = block_reduce_sum(s, sm);
    if (threadIdx.x == 0) {
        const float loss1 = ws[0] * (1.0f / (float)(B_DIM * D_DIM));
        const float loss2 = tot   * (1.0f / (float)(B_DIM * C_DIM));
        out[0] = loss1 + loss2;   // LAMDA = 1.0
    }
}

extern "C" void kernel_launch(void* const* d_in, const int* in_sizes, int n_in,
                              void* d_out, int out_size, void* d_ws, size_t ws_size,
                              hipStream_t stream) {
    const float* pos = (const float*)d_in[0];  // (B, D)
    const float* neg = (const float*)d_in[1];  // (B, C, D)
    const float* lan = (const float*)d_in[2];  // (B, D)
    float* out = (float*)d_out;                // scalar
    float* ws  = (float*)d_ws;                 // [0]=loss1 sum, [1..NBLK]=partials

    loss1_kernel<<<1, 256, 0, stream>>>(pos, lan, ws);
    loss2_kernel<<<NBLK, 256, 0, stream>>>(pos, neg, ws);
    finalize_kernel<<<1, 256, 0, stream>>>(ws, out);
}